// EfficientGlobalPointerHead_14328010900038
// MI455X (gfx1250) — compile-verified
//
#include <hip/hip_runtime.h>

// EfficientGlobalPointerHead for MI455X (gfx1250, wave32).
// Bandwidth-bound: output is 16*12*512*512 f32 = 201MB; all math (~3 GFLOP)
// is noise vs. output writes at 23.3 TB/s (~8.6us floor). Strategy:
//  - compute the head-shared q*k^T core once per 16x16 tile with
//    V_WMMA_F32_16X16X4_F32 and fan out to 12 heads in-register;
//  - projections run through a small f32 workspace (<10MB, L2-resident);
//  - proj waves carry 2 A-tiles (32 rows) so each p_w/q_w fragment feeds
//    2 WMMAs, halving L2 weight re-reads (~262MB -> ~131MB);
//  - no divergent loads in hot loops (clamped-index trick for the partial
//    q_w tile: garbage columns are never stored).

typedef __attribute__((ext_vector_type(2))) float v2f;
typedef __attribute__((ext_vector_type(8))) float v8f;

#define BB    16
#define NN    512
#define HH    1024
#define DD    64
#define HEADS 12
#define POUT  128
#define QOUT  24
#define BIGF  1000000000000.0f

static __device__ __forceinline__ v8f wmma_f32(v2f a, v2f b, v8f c) {
  return __builtin_amdgcn_wmma_f32_16x16x4_f32(
      false, a, false, b, (short)0, c, false, false);
}

// ---------------------------------------------------------------------------
// Kernel 1: outputs = hidden @ p_w^T + p_b   (8192 x 1024 x 128)
//           biasraw = (hidden @ q_w^T + q_b)/2  stored as (b, 24, n)
// One wave per 32-row tile (two 16-row A-tiles sharing every B fragment).
// WMMA f32 16x16x4 layouts: A/B lane l holds float2 at (idx=l%16, k=2*(l>>4));
// C: c[r] of lane l -> row = r + 8*(l>>4), col = l%16.
// ---------------------------------------------------------------------------
__global__ __launch_bounds__(256) void proj_kernel(
    const float* __restrict__ hidden, const float* __restrict__ p_w,
    const float* __restrict__ p_b,    const float* __restrict__ q_w,
    const float* __restrict__ q_b,
    float* __restrict__ outw, float* __restrict__ biasw)
{
  const int lane = threadIdx.x & 31;
  const int wave = threadIdx.x >> 5;
  const int tile = blockIdx.x * 8 + wave;      // 256 tiles of 32 rows
  const int r0   = tile * 32;
  const int lm   = lane & 15;
  const int kg   = lane >> 4;                  // K pair {0,1} or {2,3}
  // q tile 1 covers cols 16..31 but only 16..23 are real; clamp the address
  // (duplicated col 23 is computed but never stored) -> no exec divergence.
  const bool q1v  = (16 + lm) < QOUT;
  const int  col1 = q1v ? (16 + lm) : (QOUT - 1);

  v8f acc0[10] = {};
  v8f acc1[10] = {};

  const float* a0p = hidden + (size_t)(r0 + lm) * HH + 2 * kg;
  const float* a1p = a0p + (size_t)16 * HH;
  const float* bq0 = q_w + (size_t)lm * HH + 2 * kg;
  const float* bq1 = q_w + (size_t)col1 * HH + 2 * kg;

  for (int k = 0; k < HH; k += 4) {
    __builtin_prefetch(a0p + k + 64, 0, 3);    // global_prefetch_b8, near-cache
    v2f a0 = *(const v2f*)(a0p + k);
    v2f a1 = *(const v2f*)(a1p + k);
#pragma unroll
    for (int t = 0; t < 8; ++t) {
      v2f bv = *(const v2f*)(p_w + (size_t)(t * 16 + lm) * HH + k + 2 * kg);
      acc0[t] = wmma_f32(a0, bv, acc0[t]);
      acc1[t] = wmma_f32(a1, bv, acc1[t]);
    }
    {
      v2f bv = *(const v2f*)(bq0 + k);
      acc0[8] = wmma_f32(a0, bv, acc0[8]);
      acc1[8] = wmma_f32(a1, bv, acc1[8]);
    }
    {
      v2f bv = *(const v2f*)(bq1 + k);
      acc0[9] = wmma_f32(a0, bv, acc0[9]);
      acc1[9] = wmma_f32(a1, bv, acc1[9]);
    }
  }

  // Per-lane column biases (col depends only on (t, lm)).
  float pb[8];
#pragma unroll
  for (int t = 0; t < 8; ++t) pb[t] = p_b[t * 16 + lm];
  const float qb0 = q_b[lm];
  const float qb1 = q1v ? q_b[16 + lm] : 0.0f;

#pragma unroll
  for (int g = 0; g < 2; ++g) {
    const v8f* acc = g ? acc1 : acc0;
#pragma unroll
    for (int r = 0; r < 8; ++r) {
      const int row = r0 + g * 16 + r + 8 * kg;
      float* orow = outw + (size_t)row * POUT;
#pragma unroll
      for (int t = 0; t < 8; ++t)
        orow[t * 16 + lm] = acc[t][r] + pb[t];
      const int bidx = row >> 9;      // /512
      const int pos  = row & 511;
      biasw[((size_t)bidx * QOUT + lm) * NN + pos] = (acc[8][r] + qb0) * 0.5f;
      if (q1v)
        biasw[((size_t)bidx * QOUT + 16 + lm) * NN + pos] =
            (acc[9][r] + qb1) * 0.5f;
    }
  }
}

// ---------------------------------------------------------------------------
// Kernel 2: RoPE split. qw[d]=outputs[2d], kw[d]=outputs[2d+1]; pair i mixes
// channels (2i,2i+1) of each with ang = pos * 10000^(-2i/64).
// ---------------------------------------------------------------------------
__global__ __launch_bounds__(256) void rope_kernel(
    const float* __restrict__ outw,
    float* __restrict__ qwb, float* __restrict__ kwb)
{
  const int tid = blockIdx.x * 256 + threadIdx.x;   // B*N*32 total
  const int i   = tid & 31;
  const int row = tid >> 5;
  const int pos = row & (NN - 1);

  const float4 x = *(const float4*)(outw + (size_t)row * POUT + 4 * i);
  // 10000^(-i/32) = 2^(-i*log2(10000)/32)
  const float f   = exp2f(-(float)i * (13.2877123795494f / 32.0f));
  const float ang = (float)pos * f;
  const float s = __sinf(ang), c = __cosf(ang);

  float2 qo, ko;
  qo.x = x.x * c - x.z * s;  qo.y = x.z * c + x.x * s;   // q pair (4i, 4i+2)
  ko.x = x.y * c - x.w * s;  ko.y = x.w * c + x.y * s;   // k pair (4i+1,4i+3)
  *(float2*)(qwb + (size_t)row * DD + 2 * i) = qo;
  *(float2*)(kwb + (size_t)row * DD + 2 * i) = ko;
}

// ---------------------------------------------------------------------------
// Kernel 3: per (b, mtile, ntile): core = q*k^T via 16 WMMA steps over d=64,
// then fan out 12 heads: core/8 + biasRow[2h+1][m] + biasCol[2h][n], masks,
// strictly-lower-tri subtraction (same op order as reference).
// ---------------------------------------------------------------------------
__global__ __launch_bounds__(256) void logits_kernel(
    const float* __restrict__ qwb, const float* __restrict__ kwb,
    const float* __restrict__ biasw, const int* __restrict__ mask,
    float* __restrict__ out)
{
  __shared__ float lds[8][HEADS * 16 + 16];   // per-wave: row-bias + row-mask

  const int lane = threadIdx.x & 31;
  const int wave = threadIdx.x >> 5;
  const int tile = blockIdx.x * 8 + wave;     // 16*32*32 = 16384 tiles
  const int b    = tile >> 10;
  const int rem  = tile & 1023;
  const int m0   = (rem >> 5) * 16;
  const int n0   = (rem & 31) * 16;
  const int lm   = lane & 15;
  const int kg   = lane >> 4;

  // Stage per-row bias (odd channels) and row mask into this wave's LDS slab.
  for (int idx = lane; idx < HEADS * 16; idx += 32) {
    const int h = idx >> 4, m = idx & 15;
    lds[wave][idx] = biasw[((size_t)b * QOUT + 2 * h + 1) * NN + m0 + m];
  }
  if (lane < 16)
    lds[wave][HEADS * 16 + lane] = (float)mask[b * NN + m0 + lane];

  // Per-lane column data (even channels) in registers.
  const int gn = n0 + lm;
  float bc[HEADS];
#pragma unroll
  for (int h = 0; h < HEADS; ++h)
    bc[h] = biasw[((size_t)b * QOUT + 2 * h) * NN + gn];
  const float mc = (float)mask[b * NN + gn];

  // Head-shared 16x16 core: q(m,:) dot k(n,:) over d=64.
  v8f core = {};
  const float* qrow = qwb + ((size_t)b * NN + m0 + lm) * DD + 2 * kg;
  const float* krow = kwb + ((size_t)b * NN + n0 + lm) * DD + 2 * kg;
#pragma unroll
  for (int k = 0; k < DD; k += 4) {
    v2f a  = *(const v2f*)(qrow + k);
    v2f bv = *(const v2f*)(krow + k);
    core = wmma_f32(a, bv, core);
  }

  __syncthreads();

  const float invs = 0.125f;   // 1/sqrt(64)
#pragma unroll
  for (int r = 0; r < 8; ++r) {
    const int m  = r + 8 * kg;
    const int gm = m0 + m;
    const float mr   = lds[wave][HEADS * 16 + m];
    const float tri  = (gm > gn) ? BIGF : 0.0f;
    const float base = core[r] * invs;
#pragma unroll
    for (int h = 0; h < HEADS; ++h) {
      float v = base + lds[wave][h * 16 + m] + bc[h];
      v = (mr != 0.0f) ? v : -BIGF;   // row mask
      v = (mc != 0.0f) ? v : -BIGF;   // col mask
      v -= tri;                        // strict lower triangle
      out[(((size_t)b * HEADS + h) * NN + gm) * NN + gn] = v;
    }
  }
}

// ---------------------------------------------------------------------------
extern "C" void kernel_launch(void* const* d_in, const int* in_sizes, int n_in,
                              void* d_out, int out_size, void* d_ws, size_t ws_size,
                              hipStream_t stream) {
  const float* hidden = (const float*)d_in[0];
  const int*   mask   = (const int*)  d_in[1];
  const float* p_w    = (const float*)d_in[2];
  const float* p_b    = (const float*)d_in[3];
  const float* q_w    = (const float*)d_in[4];
  const float* q_b    = (const float*)d_in[5];
  float* out = (float*)d_out;

  char* ws = (char*)d_ws;
  float* outw  = (float*)ws;                                    // 16*512*128 f32 (4 MB)
  float* qwb   = (float*)(ws + (size_t)BB * NN * POUT * 4);     // 16*512*64  f32 (2 MB)
  float* kwb   = qwb + (size_t)BB * NN * DD;                    // 2 MB
  float* biasw = kwb + (size_t)BB * NN * DD;                    // 16*24*512 f32 (0.75 MB)

  // 8192 rows / (32 rows * 8 waves) = 32 blocks
  proj_kernel<<<32, 256, 0, stream>>>(hidden, p_w, p_b, q_w, q_b, outw, biasw);
  // 16*512*32 threads
  rope_kernel<<<(BB * NN * 32) / 256, 256, 0, stream>>>(outw, qwb, kwb);
  // 16384 tiles / 8 waves = 2048 blocks
  logits_kernel<<<(BB * 32 * 32) / 8, 256, 0, stream>>>(qwb, kwb, biasw, mask, out);
}